// DensityRatioEstimator_57088705298684
// MI455X (gfx1250) — compile-verified
//
#include <hip/hip_runtime.h>
#include <hip/hip_bf16.h>
#include <math.h>
#include <stdint.h>

typedef __attribute__((ext_vector_type(2))) float v2f;
typedef __attribute__((ext_vector_type(8))) float v8f;

#define NPTS 8192
#define DDIM 64
#define ROWS_PER_BLOCK 128                    // 8 waves * 16 rows
#define COL_CHUNKS 4
#define COLS_PER_CHUNK (NPTS / COL_CHUNKS)    // 2048
#define TILE_COLS 64                          // LDS-staged column tile
#define NTILES (COLS_PER_CHUNK / TILE_COLS)   // 32
#define LDS_STRIDE 68                         // 64 + 4 pad -> conflict-free ds_load_b64

// workspace layout (float offsets)
#define WS_ROWSUM 0             // [NPTS]  sum_j 1/(1+d2_ij)  (incl. diagonal)
#define WS_WDIAG  (NPTS)        // [NPTS]  1/(1+d2_ii)        (exact store)
#define WS_X2     (2 * NPTS)    // [NPTS]  ||z_x_j||^2
#define WS_Y2     (3 * NPTS)    // [NPTS]  ||z_y_i||^2
#define WS_SCALD  (4 * NPTS)    // 4 doubles: posL, posS, allL, allS

#define SC_POSL 0
#define SC_POSS 1
#define SC_ALLL 2
#define SC_ALLS 3

#define HAS_ASYNC_LDS  __has_builtin(__builtin_amdgcn_global_load_async_to_lds_b128)
#define HAS_WAIT_ASYNC __has_builtin(__builtin_amdgcn_s_wait_asynccnt)
#define HAS_AMD_LOG    __has_builtin(__builtin_amdgcn_logf)

// -log(1+x) for x >= 0: raw v_log_f32 (log2) * ln2, no denormal-input guard needed
__device__ __forceinline__ float neg_log1p_fast(float x) {
#if HAS_AMD_LOG
    return -0.69314718055994530942f * __builtin_amdgcn_logf(1.0f + x);
#else
    return -__logf(1.0f + x);
#endif
}

#if HAS_ASYNC_LDS
// builtin signature (from hipcc diagnostic): (AS1 int4*, AS3 int4*, imm int, imm int)
typedef int gv4i __attribute__((vector_size(16)));
typedef __attribute__((address_space(1))) gv4i glob_v4i;
typedef __attribute__((address_space(3))) gv4i lds_v4i;

// global -> LDS async copy, 16B per lane, zero data VGPRs (tracked by ASYNCcnt)
__device__ __forceinline__ void async_b128(const void* g, const void* l) {
    __builtin_amdgcn_global_load_async_to_lds_b128(
        (glob_v4i*)(uintptr_t)g,
        (lds_v4i*)(uint32_t)(uintptr_t)l,
        0, 0);
}
__device__ __forceinline__ void wait_async0() {
#if HAS_WAIT_ASYNC
    __builtin_amdgcn_s_wait_asynccnt(0);
#else
    asm volatile("s_wait_asynccnt 0x0" ::: "memory");
#endif
}
#endif

__global__ __launch_bounds__(256) void dre_init(float* __restrict__ ws) {
    int i = blockIdx.x * blockDim.x + threadIdx.x;
    if (i < NPTS) ws[WS_ROWSUM + i] = 0.0f;
    if (i < 4)    ((double*)(ws + WS_SCALD))[i] = 0.0;
}

__global__ __launch_bounds__(256) void dre_norms(const float* __restrict__ zx,
                                                 const float* __restrict__ zy,
                                                 float* __restrict__ ws) {
    int i = blockIdx.x * blockDim.x + threadIdx.x;
    const float* p = (i < NPTS) ? (zx + (size_t)i * DDIM)
                                : (zy + (size_t)(i - NPTS) * DDIM);
    float s = 0.0f;
#pragma unroll
    for (int d4 = 0; d4 < DDIM; d4 += 4) {
        float4 v = *(const float4*)(p + d4);
        s += v.x * v.x + v.y * v.y + v.z * v.z + v.w * v.w;
    }
    if (i < NPTS)            ws[WS_X2 + i] = s;
    else if (i < 2 * NPTS)   ws[WS_Y2 + (i - NPTS)] = s;
}

// Rows = z_y (A), Cols = z_x (B). Async-LDS double-buffered, fused epilogue.
__global__ __launch_bounds__(256) void dre_main(const float* __restrict__ zx,
                                                const float* __restrict__ zy,
                                                float* __restrict__ ws) {
    __shared__ float sbuf[2][TILE_COLS * LDS_STRIDE];   // 2 x 17.4KB

    const int tid  = threadIdx.x;
    const int wave = tid >> 5;
    const int lane = tid & 31;
    const int m    = lane & 15;    // M (A) / N (B,C) index
    const int h    = lane >> 4;    // K-half selector
    const int r0   = blockIdx.x * ROWS_PER_BLOCK + wave * 16;
    const int cbase = blockIdx.y * COLS_PER_CHUNK;

    // A fragments: 16x64 fp32 strip of z_y, register-resident all kernel.
    v2f aReg[16];
    const float* arow = zy + (size_t)(r0 + m) * DDIM + 2 * h;
#pragma unroll
    for (int kk = 0; kk < 16; ++kk)
        aReg[kk] = *(const v2f*)(arow + 4 * kk);

    float y2v[8];
#pragma unroll
    for (int v = 0; v < 8; ++v)
        y2v[v] = ws[WS_Y2 + r0 + v + 8 * h];

    float rowAcc[8] = {0.f, 0.f, 0.f, 0.f, 0.f, 0.f, 0.f, 0.f};
    float sPosL = 0.f, sPosS = 0.f, sAllL = 0.f, sAllS = 0.f;

    // ---- prologue: stage tile 0 into LDS buffer 0 ----
#if HAS_ASYNC_LDS
#pragma unroll
    for (int i2 = 0; i2 < 4; ++i2) {
        int f = tid + (i2 << 8), cc = f >> 4, k4 = f & 15;
        async_b128(&zx[(size_t)(cbase + cc) * DDIM + (k4 << 2)],
                   &sbuf[0][cc * LDS_STRIDE + (k4 << 2)]);
    }
    wait_async0();
#else
    float4 stg[4];
#pragma unroll
    for (int i2 = 0; i2 < 4; ++i2) {
        int f = tid + (i2 << 8), cc = f >> 4, k4 = f & 15;
        stg[i2] = *(const float4*)&zx[(size_t)(cbase + cc) * DDIM + (k4 << 2)];
    }
#pragma unroll
    for (int i2 = 0; i2 < 4; ++i2) {
        int f = tid + (i2 << 8), cc = f >> 4, k4 = f & 15;
        *(float4*)&sbuf[0][cc * LDS_STRIDE + (k4 << 2)] = stg[i2];
    }
#endif
    __syncthreads();

    for (int tt = 0; tt < NTILES; ++tt) {
        const int cur = tt & 1, nxt = cur ^ 1;
        const bool more = (tt + 1 < NTILES);

        // issue next tile's copy early; it streams into LDS under the compute phase
        if (more) {
            const int cstart = cbase + (tt + 1) * TILE_COLS;
#if HAS_ASYNC_LDS
#pragma unroll
            for (int i2 = 0; i2 < 4; ++i2) {
                int f = tid + (i2 << 8), cc = f >> 4, k4 = f & 15;
                async_b128(&zx[(size_t)(cstart + cc) * DDIM + (k4 << 2)],
                           &sbuf[nxt][cc * LDS_STRIDE + (k4 << 2)]);
            }
#else
#pragma unroll
            for (int i2 = 0; i2 < 4; ++i2) {
                int f = tid + (i2 << 8), cc = f >> 4, k4 = f & 15;
                stg[i2] = *(const float4*)&zx[(size_t)(cstart + cc) * DDIM + (k4 << 2)];
            }
#endif
        }

        // consume: 4 sub-tiles of 16 columns each
#pragma unroll
        for (int st = 0; st < 4; ++st) {
            const int c0 = cbase + tt * TILE_COLS + st * 16;
            const float* bp = &sbuf[cur][(st * 16 + m) * LDS_STRIDE + 2 * h];

            // preload all 16 B fragments (conflict-free ds_load_b64), then WMMA chain
            v2f bF[16];
#pragma unroll
            for (int kk = 0; kk < 16; ++kk)
                bF[kk] = *(const v2f*)(bp + 4 * kk);

            v8f cA = {0.f, 0.f, 0.f, 0.f, 0.f, 0.f, 0.f, 0.f};
#pragma unroll
            for (int kk = 0; kk < 16; ++kk)
                cA = __builtin_amdgcn_wmma_f32_16x16x4_f32(
                        false, aReg[kk], false, bF[kk], (short)0, cA, false, false);

            const float x2j = ws[WS_X2 + c0 + m];
            // branchless epilogue over ALL elements (diagonal included)
#pragma unroll
            for (int v = 0; v < 8; ++v) {
                float d2 = fmaxf(y2v[v] + x2j - 2.0f * cA[v], 0.0f);
                float lg = neg_log1p_fast(d2);               // -log1p(d2)
                float sg = __builtin_amdgcn_rcpf(2.0f + d2); // sigmoid(logit)
                float w  = __builtin_amdgcn_rcpf(1.0f + d2); // exp(logit)
                sAllL += lg; sAllS += sg; rowAcc[v] += w;
            }
            // diagonal sub-tile (wave-uniform, 1 of 128): extract pos terms exactly once
            if (c0 == r0) {
#pragma unroll
                for (int v = 0; v < 8; ++v) {
                    if (m == v + 8 * h) {
                        float d2 = fmaxf(y2v[v] + x2j - 2.0f * cA[v], 0.0f);
                        sPosL += neg_log1p_fast(d2);
                        sPosS += __builtin_amdgcn_rcpf(2.0f + d2);
                        ws[WS_WDIAG + r0 + v + 8 * h] = __builtin_amdgcn_rcpf(1.0f + d2);
                    }
                }
            }
        }

        if (more) {
#if HAS_ASYNC_LDS
            wait_async0();     // own wave's async copies landed; barrier covers the rest
#else
#pragma unroll
            for (int i2 = 0; i2 < 4; ++i2) {
                int f = tid + (i2 << 8), cc = f >> 4, k4 = f & 15;
                *(float4*)&sbuf[nxt][cc * LDS_STRIDE + (k4 << 2)] = stg[i2];
            }
#endif
        }
        __syncthreads();
    }

    // row sums: reduce across the 16 lanes sharing a row, one atomic per row per chunk
#pragma unroll
    for (int v = 0; v < 8; ++v) {
        float r = rowAcc[v];
        r += __shfl_xor(r, 1, 32);
        r += __shfl_xor(r, 2, 32);
        r += __shfl_xor(r, 4, 32);
        r += __shfl_xor(r, 8, 32);
        if (m == 0) atomicAdd(&ws[WS_ROWSUM + r0 + v + 8 * h], r);
    }

    // scalar sums: wave reduce in fp32, commit with fp64 atomics
    double* scal = (double*)(ws + WS_SCALD);
    float s4[4] = {sPosL, sPosS, sAllL, sAllS};
#pragma unroll
    for (int t = 0; t < 4; ++t) {
        float s = s4[t];
        s += __shfl_xor(s, 1, 32);
        s += __shfl_xor(s, 2, 32);
        s += __shfl_xor(s, 4, 32);
        s += __shfl_xor(s, 8, 32);
        s += __shfl_xor(s, 16, 32);
        if (lane == 0) atomicAdd(&scal[t], (double)s);
    }
}

__global__ __launch_bounds__(256) void dre_finalize(const float* __restrict__ ws,
                                                    float* __restrict__ out) {
    __shared__ double red[256];
    double s = 0.0;
    for (int i = threadIdx.x; i < NPTS; i += 256) {
        float rs = ws[WS_ROWSUM + i] - ws[WS_WDIAG + i];  // off-diagonal row sum
        s += (double)__logf(rs);                          // logsumexp(neg_i)
    }
    red[threadIdx.x] = s;
    __syncthreads();
    for (int off = 128; off > 0; off >>= 1) {
        if ((int)threadIdx.x < off) red[threadIdx.x] += red[threadIdx.x + off];
        __syncthreads();
    }
    if (threadIdx.x == 0) {
        const double* scal = (const double*)(ws + WS_SCALD);
        double posL = scal[SC_POSL], posS = scal[SC_POSS];
        double allL = scal[SC_ALLL], allS = scal[SC_ALLS];
        const double n  = (double)NPTS;
        const double nn = n * (n - 1.0);
        double meanPos = posL / n;
        double repulsion = red[0] / n - log(n - 1.0);
        out[0] = (float)meanPos;               // mean(pos_logits)
        out[1] = (float)((allL - posL) / nn);  // mean(neg_logits)
        out[2] = (float)(posS / n);            // mean(sigmoid(pos))
        out[3] = (float)((allS - posS) / nn);  // mean(sigmoid(neg))
        out[4] = 0.0f;                         // log_baseline
        out[5] = (float)(-meanPos + repulsion);// loss (decay term is 0)
    }
}

extern "C" void kernel_launch(void* const* d_in, const int* in_sizes, int n_in,
                              void* d_out, int out_size, void* d_ws, size_t ws_size,
                              hipStream_t stream) {
    const float* zx = (const float*)d_in[0];   // z_x (columns of logits)
    const float* zy = (const float*)d_in[1];   // z_y (rows of logits)
    float* out = (float*)d_out;
    float* ws  = (float*)d_ws;

    dre_init<<<(NPTS + 255) / 256, 256, 0, stream>>>(ws);
    dre_norms<<<(2 * NPTS + 255) / 256, 256, 0, stream>>>(zx, zy, ws);
    dim3 grid(NPTS / ROWS_PER_BLOCK, COL_CHUNKS);   // (64, 4)
    dre_main<<<grid, 256, 0, stream>>>(zx, zy, ws);
    dre_finalize<<<1, 256, 0, stream>>>(ws, out);
}